// KPlaneField_73564199846375
// MI455X (gfx1250) — compile-verified
//
#include <hip/hip_runtime.h>
#include <cstdint>
#include <cstddef>

#ifndef __has_builtin
#define __has_builtin(x) 0
#endif

#if __has_builtin(__builtin_amdgcn_global_load_async_to_lds_b32) && \
    __has_builtin(__builtin_amdgcn_s_wait_asynccnt)
#define KP_ASYNC_LDS 1
#else
#define KP_ASYNC_LDS 0
#endif

// ---------------------------------------------------------------------------
// Problem constants (from the reference)
// ---------------------------------------------------------------------------
constexpr int KP_OUT_DIM   = 32;       // channels per plane
constexpr int KP_BASE_RES  = 128;
constexpr int KP_CH_PER_PT = 288;      // 3 scales * 3 planes * 32 channels

__host__ __device__ constexpr int kp_res(int s) { return KP_BASE_RES << s; }   // 128,256,512
__host__ __device__ constexpr size_t kp_plane_elems(int s) {
    return (size_t)KP_OUT_DIM * kp_res(s) * kp_res(s);
}
// workspace layout: transposed planes [s][ci] back-to-back
__host__ __device__ constexpr size_t kp_ws_off(int s, int ci) {
    size_t o = 0;
    for (int t = 0; t < s; ++t) o += 3 * kp_plane_elems(t);
    return o + (size_t)ci * kp_plane_elems(s);
}
constexpr size_t KP_WS_ELEMS = 3 * (kp_plane_elems(0) + kp_plane_elems(1) + kp_plane_elems(2));
constexpr size_t KP_WS_BYTES = KP_WS_ELEMS * sizeof(float);

__device__ __forceinline__ int kp_imin(int a, int b) { return a < b ? a : b; }

// ---------------------------------------------------------------------------
// Kernel 1: transpose one plane (C=32, P) -> (P, C=32), P = R*R flattened (y,x).
// Block = (32,8). Each block handles a 32-channel x 32-position tile via LDS.
// Read phase uses gfx1250 async global->LDS copies when available.
// ---------------------------------------------------------------------------
__global__ __launch_bounds__(256) void kp_transpose_kernel(
    const float* __restrict__ src, float* __restrict__ dst, int P)
{
    __shared__ float tile[32][33];   // +1 pad: conflict-free transposed reads
    const int tx = threadIdx.x;      // 0..31
    const int ty = threadIdx.y;      // 0..7
    const size_t pos0 = (size_t)blockIdx.x * 32;

#pragma unroll
    for (int k = 0; k < 4; ++k) {
        const int c = ty + 8 * k;                       // channel 0..31
        const float* sp = src + (size_t)c * (size_t)P + pos0 + (size_t)tx; // coalesced in tx
#if KP_ASYNC_LDS
        // gfx1250 async copy: each lane copies 4 bytes from the global address src to the LDS address dst, tracked by ASYNCcnt.
        // Builtin signature on this toolchain uses int* pointees:
        // (as1 int* src, as3 int* dst, imm offset, imm cpol).
        // Address-space casts via integer round-trip (low 32 bits of a flat
        // LDS address are the DS offset on this HW).
        __builtin_amdgcn_global_load_async_to_lds_b32(
            (__attribute__((address_space(1))) int*)(uintptr_t)sp,
            (__attribute__((address_space(3))) int*)(uintptr_t)(&tile[c][tx]),
            /*offset=*/0, /*cpol=*/0);
#else
        tile[c][tx] = __builtin_nontemporal_load(sp);   // source is last-use
#endif
    }
#if KP_ASYNC_LDS
    __builtin_amdgcn_s_wait_asynccnt(0);
#endif
    __syncthreads();

#pragma unroll
    for (int k = 0; k < 4; ++k) {
        const int j = ty + 8 * k;                       // local position 0..31
        // lanes (tx) write consecutive channels -> coalesced 128B stores.
        // Keep RT policy: this data must stay resident in L2 for the sampler.
        dst[(pos0 + (size_t)j) * 32 + (size_t)tx] = tile[tx][j];
    }
}

// ---------------------------------------------------------------------------
// Kernel 2 (fast path): sample from channel-last planes in workspace.
// One wave per point, lane = channel. All index math is wave-uniform (scalar).
// Each corner fetch is a single fully-coalesced 128B line.
// ---------------------------------------------------------------------------
__global__ __launch_bounds__(256) void kp_sample_t_kernel(
    const float* __restrict__ pts, const float* __restrict__ aabb,
    const float* __restrict__ ws, float* __restrict__ out, int npts)
{
    const int lane = threadIdx.x & 31;
    int pt = (int)blockIdx.x * 8 + (threadIdx.x >> 5);
    pt = __builtin_amdgcn_readfirstlane(pt);        // wave-uniform -> SALU math
    if (pt >= npts) return;

    const float lo0 = aabb[0], lo1 = aabb[1], lo2 = aabb[2];
    const float hi0 = aabb[3], hi1 = aabb[4], hi2 = aabb[5];
    const size_t pb = (size_t)pt * 3;
    // q in [0,1] across the box:  x_grid = clip(q*(R-1), 0, R-1)
    float q[3];
    q[0] = (pts[pb + 0] - lo0) / (hi0 - lo0);
    q[1] = (pts[pb + 1] - lo1) / (hi1 - lo1);
    q[2] = (pts[pb + 2] - lo2) / (hi2 - lo2);

    const size_t obase = (size_t)pt * KP_CH_PER_PT + (size_t)lane;

#pragma unroll
    for (int s = 0; s < 3; ++s) {
        const int R = kp_res(s);
        const float rm1 = (float)(R - 1);
        float cx[3];
#pragma unroll
        for (int d = 0; d < 3; ++d)
            cx[d] = fminf(fmaxf(q[d] * rm1, 0.0f), rm1);

#pragma unroll
        for (int ci = 0; ci < 3; ++ci) {
            // COO pairs: (0,1), (0,2), (1,2)
            const int a = (ci == 2) ? 1 : 0;
            const int b = (ci == 0) ? 1 : 2;
            const float xf = cx[a], yf = cx[b];
            const float x0f = floorf(xf), y0f = floorf(yf);
            const float wx = xf - x0f, wy = yf - y0f;
            const int x0 = (int)x0f, y0 = (int)y0f;
            const int x1 = kp_imin(x0 + 1, R - 1);
            const int y1 = kp_imin(y0 + 1, R - 1);

            const float* tp = ws + kp_ws_off(s, ci);
            const size_t r0 = (size_t)(y0 * R);
            const size_t r1 = (size_t)(y1 * R);
            // Channel-last: one 128B line per corner, coalesced over lanes.
            const float v00 = tp[(r0 + (size_t)x0) * 32 + lane];
            const float v01 = tp[(r0 + (size_t)x1) * 32 + lane];
            const float v10 = tp[(r1 + (size_t)x0) * 32 + lane];
            const float v11 = tp[(r1 + (size_t)x1) * 32 + lane];

            const float top = v00 * (1.0f - wx) + v01 * wx;
            const float bot = v10 * (1.0f - wx) + v11 * wx;
            const float res = top * (1.0f - wy) + bot * wy;

            // Streaming 604MB output: NT store so it doesn't evict the
            // L2-resident feature planes.
            __builtin_nontemporal_store(res, out + obase + (size_t)(s * 96 + ci * 32));
        }
    }
}

// ---------------------------------------------------------------------------
// Kernel 3 (fallback, ws too small): gather straight from (C,H,W) layout.
// Same wave mapping; loads are channel-strided (uncoalesced) but correct.
// ---------------------------------------------------------------------------
__global__ __launch_bounds__(256) void kp_sample_direct_kernel(
    const float* __restrict__ pts, const float* __restrict__ aabb,
    const float* __restrict__ g0, const float* __restrict__ g1, const float* __restrict__ g2,
    const float* __restrict__ g3, const float* __restrict__ g4, const float* __restrict__ g5,
    const float* __restrict__ g6, const float* __restrict__ g7, const float* __restrict__ g8,
    float* __restrict__ out, int npts)
{
    const float* grids[9] = {g0, g1, g2, g3, g4, g5, g6, g7, g8};
    const int lane = threadIdx.x & 31;
    int pt = (int)blockIdx.x * 8 + (threadIdx.x >> 5);
    pt = __builtin_amdgcn_readfirstlane(pt);
    if (pt >= npts) return;

    const float lo0 = aabb[0], lo1 = aabb[1], lo2 = aabb[2];
    const float hi0 = aabb[3], hi1 = aabb[4], hi2 = aabb[5];
    const size_t pb = (size_t)pt * 3;
    float q[3];
    q[0] = (pts[pb + 0] - lo0) / (hi0 - lo0);
    q[1] = (pts[pb + 1] - lo1) / (hi1 - lo1);
    q[2] = (pts[pb + 2] - lo2) / (hi2 - lo2);

    const size_t obase = (size_t)pt * KP_CH_PER_PT + (size_t)lane;

#pragma unroll
    for (int s = 0; s < 3; ++s) {
        const int R = kp_res(s);
        const float rm1 = (float)(R - 1);
        float cx[3];
#pragma unroll
        for (int d = 0; d < 3; ++d)
            cx[d] = fminf(fmaxf(q[d] * rm1, 0.0f), rm1);

#pragma unroll
        for (int ci = 0; ci < 3; ++ci) {
            const int a = (ci == 2) ? 1 : 0;
            const int b = (ci == 0) ? 1 : 2;
            const float xf = cx[a], yf = cx[b];
            const float x0f = floorf(xf), y0f = floorf(yf);
            const float wx = xf - x0f, wy = yf - y0f;
            const int x0 = (int)x0f, y0 = (int)y0f;
            const int x1 = kp_imin(x0 + 1, R - 1);
            const int y1 = kp_imin(y0 + 1, R - 1);

            const float* gp = grids[s * 3 + ci] + (size_t)lane * (size_t)R * (size_t)R;
            const size_t r0 = (size_t)(y0 * R);
            const size_t r1 = (size_t)(y1 * R);
            const float v00 = gp[r0 + (size_t)x0];
            const float v01 = gp[r0 + (size_t)x1];
            const float v10 = gp[r1 + (size_t)x0];
            const float v11 = gp[r1 + (size_t)x1];

            const float top = v00 * (1.0f - wx) + v01 * wx;
            const float bot = v10 * (1.0f - wx) + v11 * wx;
            const float res = top * (1.0f - wy) + bot * wy;
            __builtin_nontemporal_store(res, out + obase + (size_t)(s * 96 + ci * 32));
        }
    }
}

// ---------------------------------------------------------------------------
// Host launcher
// ---------------------------------------------------------------------------
extern "C" void kernel_launch(void* const* d_in, const int* in_sizes, int n_in,
                              void* d_out, int out_size, void* d_ws, size_t ws_size,
                              hipStream_t stream)
{
    const float* pts  = (const float*)d_in[0];
    const float* aabb = (const float*)d_in[1];
    const float* g[9];
    for (int i = 0; i < 9; ++i) g[i] = (const float*)d_in[2 + i];
    float* out = (float*)d_out;

    const int npts    = in_sizes[0] / 3;           // 524288
    const int nblocks = (npts + 7) / 8;            // 8 waves (points) per block

    if (ws_size >= KP_WS_BYTES) {
        float* ws = (float*)d_ws;
        // Stage 1: transpose all 9 planes to channel-last layout in workspace.
        for (int s = 0; s < 3; ++s) {
            const int R = kp_res(s);
            const int P = R * R;
            for (int ci = 0; ci < 3; ++ci) {
                kp_transpose_kernel<<<P / 32, dim3(32, 8), 0, stream>>>(
                    g[s * 3 + ci], ws + kp_ws_off(s, ci), P);
            }
        }
        // Stage 2: coalesced bilinear sampling (stream order guarantees stage 1 done).
        kp_sample_t_kernel<<<nblocks, 256, 0, stream>>>(pts, aabb, ws, out, npts);
    } else {
        kp_sample_direct_kernel<<<nblocks, 256, 0, stream>>>(
            pts, aabb, g[0], g[1], g[2], g[3], g[4], g[5], g[6], g[7], g[8], out, npts);
    }
}